// ProtoSimLoss_only_prototype_65687229826059
// MI455X (gfx1250) — compile-verified
//
#include <hip/hip_runtime.h>
#include <hip/hip_bf16.h>

// ---------------- problem constants ----------------
#define BATCH      131072      // B  (2^17)
#define NVIEW      2
#define DIM        128         // D
#define GRP        128         // G = 16 subjects * 8 labels
#define ROWS_TOTAL (NVIEW * BATCH)   // 262144 rows of cf

// pass-1 (WMMA segment-sum) config
#define P1_BLOCKS  256
#define P1_ROWS    (ROWS_TOTAL / P1_BLOCKS)   // 1024 rows per block
// pass-2 (distance) config
#define P2_BLOCKS  512
#define P2_ROWS    (ROWS_TOTAL / P2_BLOCKS)   // 512 rows per block

typedef float v2f __attribute__((ext_vector_type(2)));
typedef float v8f __attribute__((ext_vector_type(8)));

// ---------------- pass 0: zero accumulators + group histogram ----------------
__global__ __launch_bounds__(256) void p0_init(float* __restrict__ centroid_sum,
                                               int* __restrict__ counts,
                                               float* __restrict__ sqrtdist_sum) {
    int i = blockIdx.x * 256 + threadIdx.x;
    if (i < GRP * DIM) centroid_sum[i] = 0.0f;
    if (i < GRP) { counts[i] = 0; sqrtdist_sum[i] = 0.0f; }
}

__global__ __launch_bounds__(256) void p0_counts(const int* __restrict__ subject,
                                                 const int* __restrict__ labels,
                                                 int* __restrict__ counts) {
    int i = blockIdx.x * 256 + threadIdx.x;
    if (i < BATCH) atomicAdd(&counts[subject[i] * 8 + labels[i]], NVIEW);
}

// ---------------- pass 1: centroid sums as one-hot WMMA GEMM ----------------
// centroid_sum[g,d] = sum_r onehot(seg(r)==g) * cf[r,d]
// C tiles [16g x 16d], K over rows (4 at a time, V_WMMA_F32_16X16X4_F32).
// Wave w owns d-columns [16w,16w+16); accumulates all 8 g-tiles in registers.
__global__ __launch_bounds__(256) void p1_centroid_wmma(const float* __restrict__ X,
                                                        const int* __restrict__ subject,
                                                        const int* __restrict__ labels,
                                                        float* __restrict__ centroid_sum) {
    const int wave = threadIdx.x >> 5;
    const int lane = threadIdx.x & 31;
    const int half = lane >> 4;     // 0 -> K rows {0,1}, 1 -> K rows {2,3}
    const int l16  = lane & 15;     // M index (A) / N index (B,C)
    const int dcol = wave * 16 + l16;
    const int rowBase = blockIdx.x * P1_ROWS;

    v8f c[8];
#pragma unroll
    for (int m = 0; m < 8; ++m) c[m] = v8f{0.f, 0.f, 0.f, 0.f, 0.f, 0.f, 0.f, 0.f};

    for (int ks = 0; ks < P1_ROWS / 4; ++ks) {
        const int r  = rowBase + ks * 4;
        const int rA = r + 2 * half;          // this lane covers K rows rA, rA+1
        const int b0 = rA & (BATCH - 1),     v0 = rA >> 17;
        const int b1 = (rA + 1) & (BATCH - 1), v1 = (rA + 1) >> 17;

        // B fragment: B[K, N=dcol]  (cf rows are contiguous 512B in X)
        v2f bf;
        bf.x = X[(b0 * NVIEW + v0) * DIM + dcol];
        bf.y = X[(b1 * NVIEW + v1) * DIM + dcol];

        const int s0 = subject[b0] * 8 + labels[b0];
        const int s1 = subject[b1] * 8 + labels[b1];

#pragma unroll
        for (int m = 0; m < 8; ++m) {
            const int g = m * 16 + l16;       // A[M=l16, K] = onehot(seg==g)
            v2f af;
            af.x = (s0 == g) ? 1.0f : 0.0f;
            af.y = (s1 == g) ? 1.0f : 0.0f;
            c[m] = __builtin_amdgcn_wmma_f32_16x16x4_f32(
                false, af, false, bf, (short)0, c[m], false, false);
        }
    }

    // C layout: VGPR j -> M = j (lanes 0-15) or j+8 (lanes 16-31), N = l16
#pragma unroll
    for (int m = 0; m < 8; ++m) {
#pragma unroll
        for (int j = 0; j < 8; ++j) {
            const int g = m * 16 + j + half * 8;
            unsafeAtomicAdd(&centroid_sum[g * DIM + dcol], c[m][j]);
        }
    }
}

// ---------------- pass 1.5: divide by counts ----------------
__global__ __launch_bounds__(256) void p15_centroid(const float* __restrict__ csum,
                                                    const int* __restrict__ counts,
                                                    float* __restrict__ centroid) {
    int i = blockIdx.x * 256 + threadIdx.x;
    if (i < GRP * DIM) centroid[i] = csum[i] / (float)counts[i >> 7];
}

// ---------------- pass 2: per-row distance, segment-sum of sqrt(dist) -------
__global__ __launch_bounds__(256) void p2_dist(const float* __restrict__ X,
                                               const int* __restrict__ subject,
                                               const int* __restrict__ labels,
                                               const float* __restrict__ centroid,
                                               float* __restrict__ sqrtdist_sum) {
    extern __shared__ float smem[];
    float* sC    = smem;              // 16384 floats: full centroid matrix
    float* sBins = smem + GRP * DIM;  // 128 bins

    for (int i = threadIdx.x; i < GRP * DIM; i += 256) sC[i] = centroid[i];
    if (threadIdx.x < GRP) sBins[threadIdx.x] = 0.0f;
    __syncthreads();

    const int wave = threadIdx.x >> 5;
    const int lane = threadIdx.x & 31;
    const int base = blockIdx.x * P2_ROWS + wave * (P2_ROWS / 8);

    for (int i = 0; i < P2_ROWS / 8; ++i) {
        const int r = base + i;
        const int b = r & (BATCH - 1);
        const int v = r >> 17;
        const int g = subject[b] * 8 + labels[b];

        const float4* row = (const float4*)(X + (b * NVIEW + v) * DIM);
        const float4* cg  = (const float4*)(sC + g * DIM);
        float4 xv = row[lane];   // dims [4*lane, 4*lane+4)
        float4 cv = cg[lane];
        float dx = xv.x - cv.x, dy = xv.y - cv.y, dz = xv.z - cv.z, dw = xv.w - cv.w;
        float ss = dx * dx + dy * dy + dz * dz + dw * dw;
#pragma unroll
        for (int off = 16; off > 0; off >>= 1) ss += __shfl_xor(ss, off, 32);
        if (lane == 0) atomicAdd(&sBins[g], sqrtf(sqrtf(ss)));  // sqrt(dist)=ss^0.25
    }
    __syncthreads();
    if (threadIdx.x < GRP)
        unsafeAtomicAdd(&sqrtdist_sum[threadIdx.x], sBins[threadIdx.x]);
}

// ---------------- pass 3: 128-element epilogue -> scalar ----------------
__global__ __launch_bounds__(128) void p3_final(const float* __restrict__ centroid,
                                                const int* __restrict__ counts,
                                                const float* __restrict__ sqrtdist_sum,
                                                float* __restrict__ out) {
    __shared__ float dens[GRP], tmp[GRP], sorted[GRP], red[GRP];
    const int t = threadIdx.x;

    const float cnt = (float)counts[t];
    float dv = (sqrtdist_sum[t] / cnt) / logf(cnt + 10.0f);
    dv = (cnt > 1.0f) ? dv : 0.0f;
    dens[t] = dv; __syncthreads();

    // max over dens (zeros included)
    red[t] = dens[t]; __syncthreads();
    for (int s = 64; s > 0; s >>= 1) { if (t < s) red[t] = fmaxf(red[t], red[t + s]); __syncthreads(); }
    const float dmax = red[0]; __syncthreads();
    dv = (cnt > 1.0f) ? dv : dmax;
    dens[t] = dv; __syncthreads();

    // rank-sort 128 values (stable), then linear-interp percentiles 10/90
    int rank = 0;
    for (int j = 0; j < GRP; ++j) {
        float vj = dens[j];
        rank += (vj < dv) || (vj == dv && j < t);
    }
    sorted[rank] = dv; __syncthreads();
    const float lo = sorted[12]  + 0.7f * (sorted[13]  - sorted[12]);   // 0.1*127 = 12.7
    const float hi = sorted[114] + 0.3f * (sorted[115] - sorted[114]);  // 0.9*127 = 114.3
    dv = fminf(fmaxf(dv, lo), hi);

    // mean + normalize
    red[t] = dv; __syncthreads();
    for (int s = 64; s > 0; s >>= 1) { if (t < s) red[t] += red[t + s]; __syncthreads(); }
    dv = 0.1f * dv / (red[0] / (float)GRP); __syncthreads();

    // cen_of_cen: thread t = dim t
    float coc = 0.0f;
    for (int g = 0; g < GRP; ++g) coc += centroid[g * DIM + t];
    tmp[t] = coc * (1.0f / (float)GRP); __syncthreads();

    // sim: thread t = group t
    float dot = 0.0f;
    for (int d = 0; d < DIM; ++d) dot += centroid[t * DIM + d] * tmp[d];
    const float sim = expf(dot / dv);

    red[t] = sim; __syncthreads();
    for (int s = 64; s > 0; s >>= 1) { if (t < s) red[t] = fmaxf(red[t], red[t + s]); __syncthreads(); }
    const float smax = red[0]; __syncthreads();
    red[t] = sim; __syncthreads();
    for (int s = 64; s > 0; s >>= 1) { if (t < s) red[t] += red[t + s]; __syncthreads(); }
    if (t == 0) out[0] = smax - red[0] / (float)GRP;   // -mean(sim - max)
}

// ---------------- launch ----------------
extern "C" void kernel_launch(void* const* d_in, const int* in_sizes, int n_in,
                              void* d_out, int out_size, void* d_ws, size_t ws_size,
                              hipStream_t stream) {
    const float* X       = (const float*)d_in[0];
    const int*   subject = (const int*)d_in[1];
    const int*   labels  = (const int*)d_in[2];
    float*       out     = (float*)d_out;

    char* ws = (char*)d_ws;
    float* centroid_sum = (float*)(ws);                       // 64 KB
    float* centroid     = (float*)(ws + 65536);               // 64 KB
    int*   counts       = (int*)  (ws + 131072);              // 512 B
    float* sqrtdist_sum = (float*)(ws + 131072 + 512);        // 512 B

    p0_init<<<(GRP * DIM + 255) / 256, 256, 0, stream>>>(centroid_sum, counts, sqrtdist_sum);
    p0_counts<<<(BATCH + 255) / 256, 256, 0, stream>>>(subject, labels, counts);
    p1_centroid_wmma<<<P1_BLOCKS, 256, 0, stream>>>(X, subject, labels, centroid_sum);
    p15_centroid<<<(GRP * DIM + 255) / 256, 256, 0, stream>>>(centroid_sum, counts, centroid);
    p2_dist<<<P2_BLOCKS, 256, (GRP * DIM + GRP) * sizeof(float), stream>>>(
        X, subject, labels, centroid, sqrtdist_sum);
    p3_final<<<1, 128, 0, stream>>>(centroid, counts, sqrtdist_sum, out);
}